// GAT_1984274890769
// MI455X (gfx1250) — compile-verified
//
#include <hip/hip_runtime.h>

typedef float v2f __attribute__((ext_vector_type(2)));
typedef float v8f __attribute__((ext_vector_type(8)));

__device__ __forceinline__ float lrelu(float v, float s) { return v > 0.f ? v : v * s; }

// float atomic max via monotonic int mapping (init location to -inf).
__device__ __forceinline__ void atomic_max_f32(float* addr, float v) {
  if (v >= 0.f) {
    atomicMax(reinterpret_cast<int*>(addr), __float_as_int(v));
  } else {
    atomicMin(reinterpret_cast<unsigned int*>(addr), __float_as_uint(v));
  }
}

// ---------------------------------------------------------------------------
// Repack a [128,128] row-major weight into K-pair-packed float2:
// WP[kp*128 + n] = (W[2kp][n], W[2kp+1][n])  -> single b64 B-fragment load.
// ---------------------------------------------------------------------------
__global__ void gat_pack_w(const float* __restrict__ W, float2* __restrict__ WP)
{
  int t = blockIdx.x * blockDim.x + threadIdx.x;   // 0 .. 64*128-1
  if (t >= 64 * 128) return;
  int kp = t >> 7, n = t & 127;
  WP[t] = make_float2(W[(2 * kp) * 128 + n], W[(2 * kp + 1) * 128 + n]);
}

// ---------------------------------------------------------------------------
// Fused dual GEMM (x@Wsrc -> XS stored, x@Wdst -> only al_d reduced) plus
// per-head attention logits al_s, al_d.
// One block = 64 rows (4 M-tiles); 8 waves, wave w owns columns [16w,16w+16)
// == head w.  B fragments are reused across the 4 M-tiles, so the K-step body
// is 4 A-loads(b64) + 2 B-loads(b64) feeding 8 V_WMMA_F32_16X16X4_F32.
// ---------------------------------------------------------------------------
template <int ACT>
__global__ __launch_bounds__(256)
void gat_gemm_attn(const float* __restrict__ X,
                   const float2* __restrict__ WsP,  // packed [64][128] float2
                   const float2* __restrict__ WdP,
                   const float* __restrict__ Asrc,  // [8,16]
                   const float* __restrict__ Adst,  // [8,16]
                   float* __restrict__ XS,          // [N,128]
                   float* __restrict__ ALS,         // [N,8]
                   float* __restrict__ ALD,         // [N,8]
                   int N)
{
  const int lane = threadIdx.x & 31;
  const int wave = threadIdx.x >> 5;      // 0..7 : column block / head
  const int m    = lane & 15;             // row within A-frag, col within B/C
  const int kh   = lane >> 4;             // half: selects K pair / M offset
  const int row0 = blockIdx.x * 64;
  const int n    = (wave << 4) + m;       // global output column 0..127

  // clamped A-row pointers (branchless; stores are guarded later)
  const float* __restrict__ xr[4];
  #pragma unroll
  for (int t = 0; t < 4; ++t) {
    int r = row0 + t * 16 + m;
    r = r < N - 1 ? r : N - 1;
    xr[t] = X + (size_t)r * 128;
  }

  v8f accS[4], accD[4];
  #pragma unroll
  for (int t = 0; t < 4; ++t) {
    accS[t] = (v8f){0.f, 0.f, 0.f, 0.f, 0.f, 0.f, 0.f, 0.f};
    accD[t] = (v8f){0.f, 0.f, 0.f, 0.f, 0.f, 0.f, 0.f, 0.f};
  }

  #pragma unroll 2
  for (int k0 = 0; k0 < 128; k0 += 4) {
    const int ka = k0 + (kh << 1);        // even; lane holds K = ka, ka+1
    const int kp = (ka >> 1) * 128 + n;
    float2 bsv = WsP[kp];
    float2 bdv = WdP[kp];
    v2f bs; bs.x = bsv.x; bs.y = bsv.y;
    v2f bd; bd.x = bdv.x; bd.y = bdv.y;

    v2f a[4];
    #pragma unroll
    for (int t = 0; t < 4; ++t) {
      float2 av = *reinterpret_cast<const float2*>(xr[t] + ka);
      if (ACT) { av.x = lrelu(av.x, 0.01f); av.y = lrelu(av.y, 0.01f); }
      a[t].x = av.x; a[t].y = av.y;
    }

    #pragma unroll
    for (int t = 0; t < 4; ++t) {
      accS[t] = __builtin_amdgcn_wmma_f32_16x16x4_f32(false, a[t], false, bs,
                                                      (short)0, accS[t], false, false);
      accD[t] = __builtin_amdgcn_wmma_f32_16x16x4_f32(false, a[t], false, bd,
                                                      (short)0, accD[t], false, false);
    }
  }

  // This wave's 16 columns are exactly head `wave`; channel c = m.
  const float asc = Asrc[(wave << 4) + m];
  const float adc = Adst[(wave << 4) + m];

  #pragma unroll
  for (int t = 0; t < 4; ++t) {
    #pragma unroll
    for (int r = 0; r < 8; ++r) {
      const int row = row0 + t * 16 + r + (kh << 3);  // C/D: M = r + 8*half
      float ps = accS[t][r] * asc;
      float pd = accD[t][r] * adc;
      #pragma unroll
      for (int mask = 8; mask >= 1; mask >>= 1) {     // 16-col reduction
        ps += __shfl_xor(ps, mask, 32);
        pd += __shfl_xor(pd, mask, 32);
      }
      if (row < N) {
        XS[(size_t)row * 128 + n] = accS[t][r];
        if (m == 0) {
          ALS[(size_t)row * 8 + wave] = ps;
          ALD[(size_t)row * 8 + wave] = pd;
        }
      }
    }
  }
}

// ---------------------------------------------------------------------------
// Per-layer init: OUT = bias broadcast, EMAX = -inf, DEN = 0.  D power of 2.
// ---------------------------------------------------------------------------
__global__ void gat_init(float* __restrict__ OUT, const float* __restrict__ bias,
                         float* __restrict__ EMAX, float* __restrict__ DEN,
                         int N, int D, int H)
{
  int t = blockIdx.x * blockDim.x + threadIdx.x;
  if (t < N * D) OUT[t] = bias[t & (D - 1)];
  if (t < N * H) { EMAX[t] = -__builtin_inff(); DEN[t] = 0.f; }
}

// ---------------------------------------------------------------------------
// Edge passes A (segment max) / B (segment exp-sum), one thread per (edge,head)
// ---------------------------------------------------------------------------
template <int H>
__global__ void gat_edge_logits(const int* __restrict__ src,
                                const int* __restrict__ dst,
                                int E, int ET,
                                const float* __restrict__ ALS,
                                const float* __restrict__ ALD,
                                float* __restrict__ EMAX,
                                float* __restrict__ DEN,
                                int phase)
{
  int t = blockIdx.x * blockDim.x + threadIdx.x;
  if (t >= ET * H) return;
  int e = t / H;
  int h = t - e * H;
  int s, d;
  if (e < E) { s = src[e]; d = dst[e]; } else { s = d = e - E; }
  float v = lrelu(ALS[(size_t)s * H + h] + ALD[(size_t)d * H + h], 0.2f);
  if (phase == 0) {
    atomic_max_f32(&EMAX[(size_t)d * H + h], v);
  } else {
    float ex = __expf(v - EMAX[(size_t)d * H + h]);
    atomicAdd(&DEN[(size_t)d * H + h], ex);
  }
}

// ---------------------------------------------------------------------------
// Edge pass C for H=8, C=16: one wave per edge. Coalesced float4 gather of the
// 128-float source row, 128 f32 atomic adds into out[dst] (hits L2).
// ---------------------------------------------------------------------------
__global__ __launch_bounds__(256)
void gat_edge_msg8(const int* __restrict__ src, const int* __restrict__ dst,
                   int E, int ET,
                   const float* __restrict__ ALS, const float* __restrict__ ALD,
                   const float* __restrict__ EMAX, const float* __restrict__ DEN,
                   const float* __restrict__ XS, float* __restrict__ OUT)
{
  int e    = (blockIdx.x * blockDim.x + threadIdx.x) >> 5;
  int lane = threadIdx.x & 31;
  if (e >= ET) return;
  int s, d;
  if (e < E) { s = src[e]; d = dst[e]; } else { s = d = e - E; }
  int h = lane >> 2;                       // 4 lanes per head, 4 channels each
  float v     = lrelu(ALS[(size_t)s * 8 + h] + ALD[(size_t)d * 8 + h], 0.2f);
  float alpha = __expf(v - EMAX[(size_t)d * 8 + h]) / DEN[(size_t)d * 8 + h];
  float4 xv = *reinterpret_cast<const float4*>(XS + (size_t)s * 128 + (lane << 2));
  float* op = OUT + (size_t)d * 128 + (lane << 2);
  atomicAdd(op + 0, xv.x * alpha);
  atomicAdd(op + 1, xv.y * alpha);
  atomicAdd(op + 2, xv.z * alpha);
  atomicAdd(op + 3, xv.w * alpha);
}

// Edge pass C for layer 3 (H=1, C=2): one thread per edge.
__global__ void gat_edge_msg2(const int* __restrict__ src, const int* __restrict__ dst,
                              int E, int ET,
                              const float* __restrict__ ALS, const float* __restrict__ ALD,
                              const float* __restrict__ EMAX, const float* __restrict__ DEN,
                              const float* __restrict__ XS3, float* __restrict__ OUT)
{
  int e = blockIdx.x * blockDim.x + threadIdx.x;
  if (e >= ET) return;
  int s, d;
  if (e < E) { s = src[e]; d = dst[e]; } else { s = d = e - E; }
  float v     = lrelu(ALS[s] + ALD[d], 0.2f);
  float alpha = __expf(v - EMAX[d]) / DEN[d];
  atomicAdd(&OUT[(size_t)d * 2 + 0], XS3[(size_t)s * 2 + 0] * alpha);
  atomicAdd(&OUT[(size_t)d * 2 + 1], XS3[(size_t)s * 2 + 1] * alpha);
}

// ---------------------------------------------------------------------------
// Layer 3 node kernel: lrelu(h2) @ w3_{src,dst} (dout=2) + attention scalars.
// ---------------------------------------------------------------------------
__global__ void gat_node3(const float* __restrict__ H2,
                          const float* __restrict__ W3s, const float* __restrict__ W3d,
                          const float* __restrict__ A3s, const float* __restrict__ A3d,
                          float* __restrict__ XS3, float* __restrict__ ALS3,
                          float* __restrict__ ALD3, int N)
{
  int i = blockIdx.x * blockDim.x + threadIdx.x;
  if (i >= N) return;
  const float* hr = H2 + (size_t)i * 128;
  float s0 = 0.f, s1 = 0.f, d0 = 0.f, d1 = 0.f;
  #pragma unroll 4
  for (int k = 0; k < 128; ++k) {
    float v = lrelu(hr[k], 0.01f);
    s0 = fmaf(v, W3s[2 * k + 0], s0);
    s1 = fmaf(v, W3s[2 * k + 1], s1);
    d0 = fmaf(v, W3d[2 * k + 0], d0);
    d1 = fmaf(v, W3d[2 * k + 1], d1);
  }
  XS3[2 * i + 0] = s0;
  XS3[2 * i + 1] = s1;
  ALS3[i] = s0 * A3s[0] + s1 * A3s[1];
  ALD3[i] = d0 * A3d[0] + d1 * A3d[1];
}

// ---------------------------------------------------------------------------
extern "C" void kernel_launch(void* const* d_in, const int* in_sizes, int n_in,
                              void* d_out, int out_size, void* d_ws, size_t ws_size,
                              hipStream_t stream)
{
  (void)n_in; (void)out_size; (void)ws_size;
  const float* x   = (const float*)d_in[0];
  const int*   ei  = (const int*)  d_in[1];
  const float* w1s = (const float*)d_in[2];
  const float* w1d = (const float*)d_in[3];
  const float* a1s = (const float*)d_in[4];
  const float* a1d = (const float*)d_in[5];
  const float* b1  = (const float*)d_in[6];
  const float* w2s = (const float*)d_in[7];
  const float* w2d = (const float*)d_in[8];
  const float* a2s = (const float*)d_in[9];
  const float* a2d = (const float*)d_in[10];
  const float* b2  = (const float*)d_in[11];
  const float* w3s = (const float*)d_in[12];
  const float* w3d = (const float*)d_in[13];
  const float* a3s = (const float*)d_in[14];
  const float* a3d = (const float*)d_in[15];
  const float* b3  = (const float*)d_in[16];

  const int N  = in_sizes[0] / 128;
  const int E  = in_sizes[1] / 2;
  const int ET = E + N;                 // with self loops
  const int* srcI = ei;
  const int* dstI = ei + E;

  // workspace (floats): xs | h | al_s | al_d | emax | den | packed weights
  float* ws   = (float*)d_ws;
  float* xs   = ws;
  float* hbuf = xs   + (size_t)128 * N;
  float* als  = hbuf + (size_t)128 * N;
  float* ald  = als  + (size_t)8 * N;
  float* emax = ald  + (size_t)8 * N;
  float* den  = emax + (size_t)8 * N;
  float2* wsP = (float2*)(den + (size_t)8 * N);   // 64*128 float2
  float2* wdP = wsP + 64 * 128;                   // 64*128 float2
  // layer-3 scalars alias the (free-by-then) al_s region
  float* xs3  = als;
  float* als3 = als + (size_t)2 * N;
  float* ald3 = als + (size_t)3 * N;

  const int TB = 256;
  dim3 blk(TB);
  dim3 gGemm((N + 63) / 64);
  dim3 gPack((64 * 128 + TB - 1) / TB);
  dim3 gInitL((N * 128 + TB - 1) / TB);
  dim3 gEdge8((ET * 8 + TB - 1) / TB);
  dim3 gMsg8((ET + 7) / 8);              // 1 wave per edge, 8 waves/block
  dim3 gEdge1((ET + TB - 1) / TB);
  dim3 gNode((N + TB - 1) / TB);
  dim3 gInit3((N * 2 + TB - 1) / TB);

  // ---- layer 1 (heads=8, ch=16) ----
  gat_pack_w<<<gPack, blk, 0, stream>>>(w1s, wsP);
  gat_pack_w<<<gPack, blk, 0, stream>>>(w1d, wdP);
  gat_gemm_attn<0><<<gGemm, blk, 0, stream>>>(x, wsP, wdP, a1s, a1d, xs, als, ald, N);
  gat_init<<<gInitL, blk, 0, stream>>>(hbuf, b1, emax, den, N, 128, 8);
  gat_edge_logits<8><<<gEdge8, blk, 0, stream>>>(srcI, dstI, E, ET, als, ald, emax, den, 0);
  gat_edge_logits<8><<<gEdge8, blk, 0, stream>>>(srcI, dstI, E, ET, als, ald, emax, den, 1);
  gat_edge_msg8<<<gMsg8, blk, 0, stream>>>(srcI, dstI, E, ET, als, ald, emax, den, xs, hbuf);

  // ---- layer 2 (lrelu(0.01) fused into GEMM A-loads) ----
  gat_pack_w<<<gPack, blk, 0, stream>>>(w2s, wsP);
  gat_pack_w<<<gPack, blk, 0, stream>>>(w2d, wdP);
  gat_gemm_attn<1><<<gGemm, blk, 0, stream>>>(hbuf, wsP, wdP, a2s, a2d, xs, als, ald, N);
  gat_init<<<gInitL, blk, 0, stream>>>(hbuf, b2, emax, den, N, 128, 8);
  gat_edge_logits<8><<<gEdge8, blk, 0, stream>>>(srcI, dstI, E, ET, als, ald, emax, den, 0);
  gat_edge_logits<8><<<gEdge8, blk, 0, stream>>>(srcI, dstI, E, ET, als, ald, emax, den, 1);
  gat_edge_msg8<<<gMsg8, blk, 0, stream>>>(srcI, dstI, E, ET, als, ald, emax, den, xs, hbuf);

  // ---- layer 3 (heads=1, ch=2) ----
  gat_node3<<<gNode, blk, 0, stream>>>(hbuf, w3s, w3d, a3s, a3d, xs3, als3, ald3, N);
  gat_init<<<gInit3, blk, 0, stream>>>((float*)d_out, b3, emax, den, N, 2, 1);
  gat_edge_logits<1><<<gEdge1, blk, 0, stream>>>(srcI, dstI, E, ET, als3, ald3, emax, den, 0);
  gat_edge_logits<1><<<gEdge1, blk, 0, stream>>>(srcI, dstI, E, ET, als3, ald3, emax, den, 1);
  gat_edge_msg2<<<gEdge1, blk, 0, stream>>>(srcI, dstI, E, ET, als3, ald3, emax, den, xs3,
                                            (float*)d_out);
}